// UpdateFunction_old_4698694222691
// MI455X (gfx1250) — compile-verified
//
#include <hip/hip_runtime.h>
#include <math.h>

typedef __attribute__((ext_vector_type(2))) float v2f;
typedef __attribute__((ext_vector_type(8))) float v8f;

// ---------------- pair metadata (l1 outer asc, l2 inner asc; p = l1(l1+1)/2 + l2) ----
__constant__ int c_P_L1[10] = {0,1,1,2,2,2,3,3,3,3};
__constant__ int c_P_L2[10] = {0,0,1,0,1,2,0,1,2,3};
// flat offsets of each pair's CG block inside the table built by cg_init_kernel
__constant__ int c_CG_OFF[10] = {0,1,10,91,116,341,741,790,1042,1567}; // total 2351 floats
// position of pair p inside output-l's concatenation list (reference loop order)
__constant__ int c_POS[10][4] = {
  {0,0,0,0},  // (0,0): l0 pos0
  {0,0,0,0},  // (1,0): l1 pos0
  {1,1,0,0},  // (1,1)
  {0,0,1,0},  // (2,0)
  {0,2,2,0},  // (2,1)
  {2,3,3,1},  // (2,2)
  {0,0,0,2},  // (3,0)
  {0,0,4,3},  // (3,1)
  {0,4,5,4},  // (3,2)
  {3,5,6,5}   // (3,3)
};
// output-l flat base offsets in d_out (floats) and channel counts n_l*1024
__constant__ unsigned c_BASE[4] = {0u, 2097152u, 11534336u, 29884416u};
__constant__ int c_NCH[4] = {4096, 6144, 7168, 6144};

// ---------------- CG coefficient init (runs once per launch; deterministic) ----------
__device__ double dfact(int n) { double r = 1.0; for (int i = 2; i <= n; ++i) r *= (double)i; return r; }

__device__ double cg_coeff(int l1, int l2, int l, int m1, int m2, int m) {
  if (m1 + m2 != m) return 0.0;
  double pre = sqrt((2.0 * l + 1.0) * dfact(l1 + l2 - l) * dfact(l1 - l2 + l) *
                    dfact(-l1 + l2 + l) / dfact(l1 + l2 + l + 1));
  pre *= sqrt(dfact(l + m) * dfact(l - m) * dfact(l1 - m1) * dfact(l1 + m1) *
              dfact(l2 - m2) * dfact(l2 + m2));
  int kmin = 0;
  if (l2 - l - m1 > kmin) kmin = l2 - l - m1;
  if (l1 + m2 - l > kmin) kmin = l1 + m2 - l;
  int kmax = l1 + l2 - l;
  if (l1 - m1 < kmax) kmax = l1 - m1;
  if (l2 + m2 < kmax) kmax = l2 + m2;
  double s = 0.0;
  for (int k = kmin; k <= kmax; ++k) {
    double d = dfact(k) * dfact(l1 + l2 - l - k) * dfact(l1 - m1 - k) *
               dfact(l2 + m2 - k) * dfact(l - l2 + m1 + k) * dfact(l - l1 - m2 + k);
    s += (k & 1) ? (-1.0 / d) : (1.0 / d);
  }
  return pre * s;
}

__global__ void cg_init_kernel(float* __restrict__ cg) {
  int p = blockIdx.x;
  int l1 = c_P_L1[p], l2 = c_P_L2[p];
  int M1 = 2 * l1 + 1, M2 = 2 * l2 + 1;
  int lmin = l1 - l2;
  int lmx = (l1 + l2 < 3) ? (l1 + l2) : 3;
  int nlm = 0;
  for (int l = lmin; l <= lmx; ++l) nlm += 2 * l + 1;
  int tot = nlm * M1 * M2;
  int off = c_CG_OFF[p];
  for (int idx = threadIdx.x; idx < tot; idx += blockDim.x) {
    int k  = idx % (M1 * M2);
    int lm = idx / (M1 * M2);
    int l = lmin, rem = lm;
    while (rem >= 2 * l + 1) { rem -= 2 * l + 1; ++l; }
    int m  = rem - l;
    int m1 = k / M2 - l1;
    int m2 = k % M2 - l2;
    cg[off + idx] = (float)cg_coeff(l1, l2, l, m1, m2, m);
  }
}

// ---------------- main templated kernel per (l1,l2) pair -----------------------------
constexpr int nlm_of(int lmin, int lmx) { int s = 0; for (int l = lmin; l <= lmx; ++l) s += 2 * l + 1; return s; }

template <int L1, int L2>
__global__ __launch_bounds__(256) void cgprod_kernel(const float* __restrict__ fa,
                                                     const float* __restrict__ fb,
                                                     const float* __restrict__ cg,
                                                     float* __restrict__ out) {
  constexpr int M1    = 2 * L1 + 1;
  constexpr int M2    = 2 * L2 + 1;
  constexpr int LMIN  = L1 - L2;
  constexpr int LMX   = (L1 + L2 < 3) ? (L1 + L2) : 3;
  constexpr int NLM   = nlm_of(LMIN, LMX);     // total (l,m) output rows
  constexpr int NCOL  = 2 * NLM;               // real/imag interleaved columns
  constexpr int K2    = 2 * M1;                // GEMM K (ar|ai)
  constexpr int KST   = (K2 + 3) / 4;          // WMMA k-steps of 4
  constexpr int KP    = KST * 4;               // padded K (<=16)
  constexpr int NCT   = (NCOL + 15) / 16;      // 16-wide column tiles
  constexpr int NCOLP = NCT * 16;              // padded columns
  constexpr int PAIR  = L1 * (L1 + 1) / 2 + L2;
  constexpr int CGSZ  = NLM * M1 * M2;

  __shared__ float sa[2][M1][32];                       // [re/im][m1][t1]
  __shared__ float sb[2][M2][32];                       // [re/im][m2][t2]
  __shared__ float sCG[CGSZ];                           // CG block for this pair
  __shared__ alignas(16) float Apad[32][16];            // zero-padded A panel [t1][k]
  __shared__ alignas(16) float Bm[16][NCOLP][16];       // B panel for 16 t2 [t2l][n][k]
  __shared__ unsigned sOff[NCOLP];                      // per-column out base (floats)
  __shared__ unsigned sStr[NCOLP];                      // per-column stride = (2l+1)*2

  const int b   = blockIdx.x;
  const int tid = threadIdx.x;

  // ---- stage inputs: f[b] layout is [M][32][2] contiguous; plus CG block
  const float* pa = fa + (size_t)b * (M1 * 64);
  for (int i = tid; i < M1 * 64; i += 256) {
    int m = i >> 6, r = i & 63;
    sa[r & 1][m][r >> 1] = pa[i];
  }
  const float* pb = fb + (size_t)b * (M2 * 64);
  for (int i = tid; i < M2 * 64; i += 256) {
    int m = i >> 6, r = i & 63;
    sb[r & 1][m][r >> 1] = pb[i];
  }
  const float* cgp = cg + c_CG_OFF[PAIR];
  for (int i = tid; i < CGSZ; i += 256) sCG[i] = cgp[i];
  __syncthreads();

  // ---- zero-padded A panel: Apad[t][k] = k<M1 ? ar[k][t] : (k<2M1 ? ai[k-M1][t] : 0)
  for (int i = tid; i < 32 * 16; i += 256) {
    int k = i >> 5, t = i & 31;                 // k slow -> uniform per 32-thread group
    float v = 0.0f;
    if (k < M1) v = sa[0][k][t];
    else if (k < K2) v = sa[1][k - M1][t];
    Apad[t][k] = v;
  }

  // ---- per-column output address tables
  if (tid < NCOLP) {
    int n = tid;
    unsigned off = 0, st = 0;
    if (n < NCOL) {
      int lm = n >> 1, c = n & 1;
      int l = LMIN, rem = lm;
      while (rem >= 2 * l + 1) { rem -= 2 * l + 1; ++l; }
      int m = rem;
      int pos = c_POS[PAIR][l];
      st = (unsigned)((2 * l + 1) * 2);
      off = c_BASE[l] + ((unsigned)b * (unsigned)c_NCH[l] + (unsigned)pos * 1024u) * st
            + (unsigned)(m * 2 + c);
    }
    sOff[n] = off;
    sStr[n] = st;
  }

  const int lane  = tid & 31;
  const int wave  = tid >> 5;
  const int laneM = lane & 15;   // A: row, B/D: column within tile
  const int laneH = lane >> 4;   // selects K-pair (A/B) or row-half (D)

  // ---- process t2 in two halves of 16 to bound LDS
  for (int h = 0; h < 2; ++h) {
    __syncthreads();   // Apad/tables ready (h=0); previous GEMM done (h=1)

    // B panel: Bm[t2l][n][k] with complex packing and signs baked in, zero padded.
    // col n=(lm,c): k<M1 -> (c? Hi : Hr)[lm][k],  M1<=k<2M1 -> (c? Hr : -Hi)[lm][k-M1]
    // H{r,i}[lm][m1][t2] = sum_m2 CG[lm][m1*M2+m2] * sb[{0,1}][m2][t2]
    for (int i = tid; i < 16 * NCOLP * 16; i += 256) {
      int t2l = i & 15;
      int j   = i >> 4;
      int n   = j % NCOLP;
      int k   = j / NCOLP;                      // k slow -> uniform per group
      int t2  = h * 16 + t2l;
      float val = 0.0f;
      if (k < K2 && n < NCOL) {
        int lm = n >> 1, c = n & 1;
        int m1  = (k < M1) ? k : (k - M1);
        int src = (k < M1) ? c : (c ^ 1);       // which of br/bi feeds this element
        float s = 0.0f;
#pragma unroll
        for (int m2 = 0; m2 < M2; ++m2)
          s += sCG[lm * (M1 * M2) + m1 * M2 + m2] * sb[src][m2][t2];
        val = (k >= M1 && c == 0) ? -s : s;
      }
      Bm[t2l][n][k] = val;
    }
    __syncthreads();

    // ---- GEMM: D[32 x NCOLP] = Apad * Bm per t2; pure b64 LDS loads + WMMA
    constexpr int NTASK = 16 * 2 * NCT;         // (t2l, rowtile, coltile); mult of 8
    for (int task = wave; task < NTASK; task += 8) {
      const int ct  = task % NCT;
      const int rt  = (task / NCT) & 1;
      const int t2l = task / (2 * NCT);
      const int row = rt * 16 + laneM;
      const int n   = ct * 16 + laneM;

      v8f acc = {};
#pragma unroll
      for (int kk = 0; kk < KST; ++kk) {
        v2f af = *(const v2f*)&Apad[row][kk * 4 + laneH * 2];
        v2f bf = *(const v2f*)&Bm[t2l][n][kk * 4 + laneH * 2];
        acc = __builtin_amdgcn_wmma_f32_16x16x4_f32(false, af, false, bf,
                                                    (short)0, acc, false, false);
      }

      // ---- scatter D tile: lane holds column n, rows rt*16 + laneH*8 + v
      if (n < NCOL) {
        const unsigned st = sStr[n];
        const int t2 = h * 16 + t2l;
        unsigned idx = sOff[n] + (unsigned)((rt * 16 + laneH * 8) * 32 + t2) * st;
#pragma unroll
        for (int v = 0; v < 8; ++v) {
          out[idx] = acc[v];
          idx += 32u * st;
        }
      }
    }
  }
}

// ---------------- host launcher ------------------------------------------------------
extern "C" void kernel_launch(void* const* d_in, const int* in_sizes, int n_in,
                              void* d_out, int out_size, void* d_ws, size_t ws_size,
                              hipStream_t stream) {
  (void)in_sizes; (void)n_in; (void)out_size; (void)ws_size;
  float* cgtab = (float*)d_ws;          // 2351 floats of CG coefficients
  float* outp  = (float*)d_out;

  cg_init_kernel<<<dim3(10), dim3(128), 0, stream>>>(cgtab);

#define LAUNCH_PAIR(A, B2)                                                       \
  cgprod_kernel<A, B2><<<dim3(256), dim3(256), 0, stream>>>(                     \
      (const float*)d_in[A], (const float*)d_in[B2], cgtab, outp)

  LAUNCH_PAIR(0, 0);
  LAUNCH_PAIR(1, 0);
  LAUNCH_PAIR(1, 1);
  LAUNCH_PAIR(2, 0);
  LAUNCH_PAIR(2, 1);
  LAUNCH_PAIR(2, 2);
  LAUNCH_PAIR(3, 0);
  LAUNCH_PAIR(3, 1);
  LAUNCH_PAIR(3, 2);
  LAUNCH_PAIR(3, 3);
#undef LAUNCH_PAIR
}